// Bilinear_18691697672581
// MI455X (gfx1250) — compile-verified
//
#include <hip/hip_runtime.h>

typedef __attribute__((ext_vector_type(2))) float v2f;
typedef __attribute__((ext_vector_type(8))) float v8f;

#define HDIM 1024
#define BDIM 8
// rows per batch = N*I = 64*128 = 8192  (power of two: row>>13 = b)

// ---------------------------------------------------------------------------
// Kernel 1: v[b][j] = sum_k W[j][k] * hidden[b][k]
// One wave32 per 16-row j-tile, V_WMMA_F32_16X16X4_F32, batch padded 8->16
// with a zero mask (branchless so EXEC stays all-1s for the WMMA).
// ---------------------------------------------------------------------------
__global__ void __launch_bounds__(32)
bilinear_wv_wmma(const float* __restrict__ W,
                 const float* __restrict__ hidden,
                 float* __restrict__ v)
{
    const int lane = threadIdx.x;        // 0..31
    const int j0   = blockIdx.x * 16;    // 64 blocks cover j=0..1023
    const int m    = lane & 15;          // A row within tile
    const int kp   = (lane >> 4) * 2;    // K sub-offset: 0 or 2
    const int n    = lane & 15;          // B/D column = (padded) batch index

    const float* Arow = W + (size_t)(j0 + m) * HDIM + kp;
    const float* Brow = hidden + (size_t)((n < BDIM) ? n : 0) * HDIM + kp;
    const float  bmask = (n < BDIM) ? 1.0f : 0.0f;

    v8f acc0 = {};
    v8f acc1 = {};
    #pragma unroll 2
    for (int k0 = 0; k0 < HDIM; k0 += 8) {
        // first 16x16x4 slab
        v2f a0 = *(const v2f*)(Arow + k0);
        v2f h0 = *(const v2f*)(Brow + k0);
        v2f b0; b0.x = h0.x * bmask; b0.y = h0.y * bmask;
        acc0 = __builtin_amdgcn_wmma_f32_16x16x4_f32(false, a0, false, b0,
                                                     (short)0, acc0, false, false);
        // second 16x16x4 slab (independent accumulator breaks RAW chain)
        v2f a1 = *(const v2f*)(Arow + k0 + 4);
        v2f h1 = *(const v2f*)(Brow + k0 + 4);
        v2f b1; b1.x = h1.x * bmask; b1.y = h1.y * bmask;
        acc1 = __builtin_amdgcn_wmma_f32_16x16x4_f32(false, a1, false, b1,
                                                     (short)0, acc1, false, false);
    }
    v8f acc = acc0 + acc1;

    if (n < BDIM) {
        const int mbase = (lane >> 4) * 8;   // D: VGPR g -> M = g + 8*(lane/16)
        #pragma unroll
        for (int g = 0; g < 8; ++g)
            v[(size_t)n * HDIM + j0 + mbase + g] = acc[g];
    }
}

// ---------------------------------------------------------------------------
// Kernel 2 (bandwidth-bound main op): out[row] = enc[row,:] . v[b,:] + bias
// One wave32 per row; float4 global loads along H (fully coalesced per wave),
// v[b,:] staged in LDS, butterfly shuffle reduction.
// ---------------------------------------------------------------------------
__global__ void __launch_bounds__(256)
bilinear_dot(const float* __restrict__ enc,
             const float* __restrict__ v,
             const float* __restrict__ bias,
             float* __restrict__ out)
{
    __shared__ float sv[HDIM];
    const int t = threadIdx.x;
    const int b = (blockIdx.x * 8) >> 13;        // 8192 rows per batch; block never straddles b

    // Stage v[b,:] (4 KB) into LDS: 256 threads x float4
    ((float4*)sv)[t] = ((const float4*)(v + (size_t)b * HDIM))[t];
    __syncthreads();

    const int    wave = t >> 5;
    const int    lane = t & 31;
    const size_t row  = (size_t)blockIdx.x * 8 + wave;
    const float* erow = enc + row * HDIM;

    float4 acc = make_float4(0.f, 0.f, 0.f, 0.f);
    #pragma unroll
    for (int c = 0; c < 8; ++c) {
        const int k  = c * 128 + lane * 4;       // wave covers 128 floats per step
        float4 e  = *(const float4*)(erow + k);
        float4 vv = *(const float4*)(sv + k);
        acc.x += e.x * vv.x;
        acc.y += e.y * vv.y;
        acc.z += e.z * vv.z;
        acc.w += e.w * vv.w;
    }
    float s = (acc.x + acc.y) + (acc.z + acc.w);
    #pragma unroll
    for (int off = 16; off > 0; off >>= 1)
        s += __shfl_xor(s, off, 32);

    if (lane == 0)
        out[row] = s + bias[0];
}

// ---------------------------------------------------------------------------
extern "C" void kernel_launch(void* const* d_in, const int* in_sizes, int n_in,
                              void* d_out, int out_size, void* d_ws, size_t ws_size,
                              hipStream_t stream)
{
    // setup_inputs order: hidden, encoder_hiddens, input_lengths, W, b
    const float* hidden = (const float*)d_in[0];
    const float* enc    = (const float*)d_in[1];
    // d_in[2] = input_lengths (unused by the module's math)
    const float* W      = (const float*)d_in[3];
    const float* bias   = (const float*)d_in[4];

    float* v = (float*)d_ws;                     // 8*1024 f32 = 32 KB scratch

    // Step 1: v = W @ hidden^T  (tiny GEMM, WMMA f32 path)
    bilinear_wv_wmma<<<HDIM / 16, 32, 0, stream>>>(W, hidden, v);

    // Step 2: streaming batched GEMV (256 MB @ ~23.3 TB/s -> ~11 us floor)
    // 65536 rows, 8 rows (8 waves) per block
    bilinear_dot<<<65536 / 8, 256, 0, stream>>>(enc, v, bias, (float*)d_out);
}